// RPN_13116830121991
// MI455X (gfx1250) — compile-verified
//
#include <hip/hip_runtime.h>
#include <hip/hip_bf16.h>
#include <cmath>

// ---------------------------------------------------------------------------
// RPN (Faster R-CNN region proposal network) for MI455X / gfx1250.
// Heavy math (3x3 conv, 1x1 heads) runs on the matrix cores via
// V_WMMA_F32_16X16X4_F32 (fp32 in/out -> numerics faithful to the fp32
// reference; argsort+NMS downstream is discontinuous in the scores, so we do
// NOT downcast to bf16). Weight tiles are staged into LDS with the CDNA5
// async DMA path (GLOBAL_LOAD_ASYNC_TO_LDS_B32 + s_wait_asynccnt) and the
// K-loop is double-buffered so staging overlaps the WMMA stream.
// ---------------------------------------------------------------------------

#define RPN_H 64
#define RPN_W 100
#define RPN_M 6400          // H*W spatial positions
#define RPN_IC 1024
#define RPN_OC 512
#define RPN_A 9
#define NBOX (RPN_M * RPN_A)     // 57600
#define NPAD 65536               // next pow2 for bitonic sort
#define PRE_N 6000
#define POST_N 300
#define NMS_WORDS 188            // ceil(6000/32)

typedef float v2f __attribute__((ext_vector_type(2)));
typedef float v8f __attribute__((ext_vector_type(8)));

struct Anchors { float a[RPN_A * 4]; };

// ---------------------------------------------------------------------------
// Kernel 1: 3x3 conv (pad 1) + bias + ReLU as 9 accumulated implicit GEMMs.
// M = 6400 spatial, N = 512 out-channels, K = 1024 in-channels per tap.
// Block: 256 threads = 8 waves arranged 4(M) x 2(N); block tile 64M x 64N.
// K staged in chunks of 16 through double-buffered LDS, padded stride 22
// (even for b64 alignment; 22*ln mod 64 distinct over 32 lanes -> conflict-
// free fragment reads). WMMA f32 16x16x4 fragment layout (ISA 7.12.2):
//   lane L holds A[M = L%16][K = 2*(L/16) + v] -> one 8-byte LDS read each.
// B tiles (no zero-fill needed) go global->LDS via async DMA (ASYNCcnt);
// A tiles (need halo zero-fill) use predicated loads + ds_store.
// ---------------------------------------------------------------------------
__global__ __launch_bounds__(256) void rpn_conv3x3_wmma(
    const float* __restrict__ feat,   // [1024][64][100]
    const float* __restrict__ wgt,    // [512][1024][3][3]
    const float* __restrict__ bias,   // [512]
    float* __restrict__ out)          // [6400][512]  (m-major, relu'ed)
{
  const int KC = 16, LDA = 22, LDB = 22;
  __shared__ float As[2][64 * LDA];   // [buf][m_local][k]
  __shared__ float Bs[2][64 * LDB];   // [buf][n_local][k]

  const int bm = blockIdx.x * 64;
  const int bn = blockIdx.y * 64;
  const int t    = threadIdx.x;
  const int lane = t & 31;
  const int wv   = t >> 5;
  const int wm   = wv & 3;     // 0..3 : M sub-tile
  const int wn   = wv >> 2;    // 0..1 : N sub-tile (each wave does 16M x 32N)
  const int ln   = lane & 15;
  const int hi   = lane >> 4;

  v8f acc0 = {};
  v8f acc1 = {};

  const int mi = t & 63;       // loader row (m or n local)
  const int kb = t >> 6;       // loader k base (0..3)

  // Precomputed per-thread spatial coords for the A loader.
  const int m_ld = bm + mi;
  const int h_ld = m_ld / RPN_W;
  const int w_ld = m_ld - h_ld * RPN_W;

  // Stage chunk c (c = tap*64 + kchunk) into LDS buffer `buf`.
  auto stage = [&](int c, int buf) {
    const int tap = c >> 6;
    const int kc  = (c & 63) * KC;
    const int dh  = tap / 3 - 1;
    const int dw  = tap % 3 - 1;
    const int hh  = h_ld + dh;
    const int ww  = w_ld + dw;
    const bool inb = (unsigned)hh < (unsigned)RPN_H &&
                     (unsigned)ww < (unsigned)RPN_W;
    const float* ap = feat + (size_t)(kc + kb) * RPN_M + hh * RPN_W + ww;
    const float* bp = wgt + ((size_t)(bn + mi) * RPN_IC + (kc + kb)) * 9 + tap;
#pragma unroll
    for (int r = 0; r < 4; ++r) {
      // A: predicated (halo zero-fill) -> ds_store
      float av = 0.f;
      if (inb) av = ap[(size_t)r * 4 * RPN_M];
      As[buf][mi * LDA + kb + r * 4] = av;
      // B: async DMA straight into LDS (CDNA5 GLOBAL_LOAD_ASYNC_TO_LDS_B32,
      // tracked with ASYNCcnt; no VGPR round-trip).
      const unsigned lds_off =
          (unsigned)(size_t)&Bs[buf][mi * LDB + kb + r * 4];
      const unsigned long long gaddr =
          (unsigned long long)(size_t)(bp + (size_t)r * 4 * 9);
      asm volatile("global_load_async_to_lds_b32 %0, %1, off"
                   :: "v"(lds_off), "v"(gaddr) : "memory");
    }
  };

  const int NC = 9 * (RPN_IC / KC);   // 576 chunks
  stage(0, 0);
  for (int c = 0; c < NC; ++c) {
    const int buf = c & 1;
    // Our async writes (and, after the barrier, everyone's) are complete.
    asm volatile("s_wait_asynccnt 0x0" ::: "memory");
    __syncthreads();
    if (c + 1 < NC) stage(c + 1, buf ^ 1);   // prefetch next chunk

    const float* a_base = &As[buf][(wm * 16 + ln) * LDA];
    const float* b_base = &Bs[buf][(wn * 32 + ln) * LDB];
#pragma unroll
    for (int ks = 0; ks < 4; ++ks) {
      const int k0 = ks * 4 + 2 * hi;
      v2f af = *(const v2f*)(a_base + k0);
      v2f b0 = *(const v2f*)(b_base + k0);
      v2f b1 = *(const v2f*)(b_base + 16 * LDB + k0);
      acc0 = __builtin_amdgcn_wmma_f32_16x16x4_f32(
          false, af, false, b0, (short)0, acc0, false, false);
      acc1 = __builtin_amdgcn_wmma_f32_16x16x4_f32(
          false, af, false, b1, (short)0, acc1, false, false);
    }
  }

  // ---- epilogue: bias + ReLU, store m-major (k-contiguous for the heads) ----
  const int n0 = bn + wn * 32 + ln;
  const float bia0 = bias[n0];
  const float bia1 = bias[n0 + 16];
#pragma unroll
  for (int r = 0; r < 8; ++r) {
    const int m = bm + wm * 16 + r + 8 * hi;   // C layout: VGPR r -> M=r / r+8
    float v0 = acc0[r] + bia0;  v0 = v0 > 0.f ? v0 : 0.f;
    float v1 = acc1[r] + bia1;  v1 = v1 > 0.f ? v1 : 0.f;
    out[(size_t)m * RPN_OC + n0]      = v0;
    out[(size_t)m * RPN_OC + n0 + 16] = v1;
  }
}

// ---------------------------------------------------------------------------
// Kernel 2: fused 1x1 heads. N = 64 (18 cls + 36 reg + 10 zero pad), K = 512.
// feats is [m][k] k-contiguous and weights are [n][k] k-contiguous, so both
// fragments are direct 8-byte global loads -- no LDS needed (all hot data
// lives in the 192MB L2 anyway).
// ---------------------------------------------------------------------------
__global__ __launch_bounds__(256) void rpn_heads_wmma(
    const float* __restrict__ feats,  // [6400][512]
    const float* __restrict__ clsw,   // [18][512]
    const float* __restrict__ clsb,   // [18]
    const float* __restrict__ regw,   // [36][512]
    const float* __restrict__ regb,   // [36]
    float* __restrict__ head)         // [6400][64]
{
  const int t    = threadIdx.x;
  const int lane = t & 31;
  const int wv   = t >> 5;
  const int wm   = wv >> 2;    // 0..1
  const int wn   = wv & 3;     // 0..3
  const int ln   = lane & 15;
  const int hi   = lane >> 4;

  const int mb = blockIdx.x * 32 + wm * 16;
  const int n  = wn * 16 + ln;

  const float* wrow = nullptr;
  float bia = 0.f;
  if (n < 18)      { wrow = clsw + (size_t)n * 512;        bia = clsb[n]; }
  else if (n < 54) { wrow = regw + (size_t)(n - 18) * 512; bia = regb[n - 18]; }

  const float* arow = feats + (size_t)(mb + ln) * 512;

  v8f acc = {};
  for (int k = 0; k < 512; k += 4) {
    const int k0 = k + 2 * hi;
    v2f af = *(const v2f*)(arow + k0);
    v2f bf = {0.f, 0.f};
    if (wrow) bf = *(const v2f*)(wrow + k0);
    acc = __builtin_amdgcn_wmma_f32_16x16x4_f32(
        false, af, false, bf, (short)0, acc, false, false);
  }
#pragma unroll
  for (int r = 0; r < 8; ++r)
    head[(size_t)(mb + r + 8 * hi) * 64 + n] = acc[r] + bia;
}

// ---------------------------------------------------------------------------
// Kernel 3: proposal decode. Per anchor: 2-way softmax (== sigmoid of logit
// diff), bbox transform, clip, min-size filter, and a 64-bit sort key
// (order-preserving float bits << 32) | (~index) so a descending key sort
// reproduces jnp's stable argsort(-scores) tie-breaking exactly.
// ---------------------------------------------------------------------------
__global__ __launch_bounds__(256) void rpn_proposals(
    const float* __restrict__ head, const float* __restrict__ im_info,
    Anchors an, float* __restrict__ boxes, int* __restrict__ valid,
    unsigned long long* __restrict__ keys)
{
  const int i = blockIdx.x * 256 + threadIdx.x;
  if (i >= NPAD) return;
  if (i >= NBOX) { keys[i] = 0ull; return; }   // pad sorts last (descending)

  const int m = i / RPN_A;
  const int a = i - m * RPN_A;
  const int h = m / RPN_W;
  const int w = m - h * RPN_W;
  const float* hm = head + (size_t)m * 64;

  const float s0 = hm[a];            // bg logit
  const float s1 = hm[RPN_A + a];    // fg logit
  const float score = 1.f / (1.f + __expf(s0 - s1));

  const float dx = hm[18 + 4 * a + 0];
  const float dy = hm[18 + 4 * a + 1];
  const float dw = hm[18 + 4 * a + 2];
  const float dh = hm[18 + 4 * a + 3];

  const float sx = (float)w * 16.f, sy = (float)h * 16.f;
  const float ax1 = an.a[a * 4 + 0] + sx, ay1 = an.a[a * 4 + 1] + sy;
  const float ax2 = an.a[a * 4 + 2] + sx, ay2 = an.a[a * 4 + 3] + sy;
  const float aw = ax2 - ax1 + 1.f, ah = ay2 - ay1 + 1.f;
  const float cx = ax1 + 0.5f * aw, cy = ay1 + 0.5f * ah;

  const float pcx = dx * aw + cx, pcy = dy * ah + cy;
  const float pw  = __expf(dw) * aw, ph = __expf(dh) * ah;

  const float imh = im_info[0], imw = im_info[1];
  const float x1 = fminf(fmaxf(pcx - 0.5f * pw, 0.f), imw - 1.f);
  const float y1 = fminf(fmaxf(pcy - 0.5f * ph, 0.f), imh - 1.f);
  const float x2 = fminf(fmaxf(pcx + 0.5f * pw, 0.f), imw - 1.f);
  const float y2 = fminf(fmaxf(pcy + 0.5f * ph, 0.f), imh - 1.f);

  const bool vld = (x2 - x1 + 1.f >= 16.f) && (y2 - y1 + 1.f >= 16.f);
  const float sc = vld ? score : -1.f;

  boxes[i * 4 + 0] = x1; boxes[i * 4 + 1] = y1;
  boxes[i * 4 + 2] = x2; boxes[i * 4 + 3] = y2;
  valid[i] = vld ? 1 : 0;

  unsigned u = __float_as_uint(sc);
  u = (u >> 31) ? ~u : (u | 0x80000000u);        // order-preserving transform
  keys[i] = ((unsigned long long)u << 32) | (0xFFFFFFFFu - (unsigned)i);
}

// ---------------------------------------------------------------------------
// Kernel 4: one bitonic compare-exchange pass (descending). 136 passes total.
// ---------------------------------------------------------------------------
__global__ __launch_bounds__(256) void rpn_bitonic(
    unsigned long long* __restrict__ keys, int j, int k)
{
  const int i = blockIdx.x * 256 + threadIdx.x;
  const int ixj = i ^ j;
  if (ixj > i) {
    unsigned long long a = keys[i], b = keys[ixj];
    const bool up = ((i & k) == 0);              // up-block wants larger first
    if (up ? (a < b) : (a > b)) { keys[i] = b; keys[ixj] = a; }
  }
}

// Kernel 5: gather top PRE_N boxes/valid in rank order.
__global__ __launch_bounds__(256) void rpn_gather_top(
    const unsigned long long* __restrict__ keys,
    const float* __restrict__ boxes, const int* __restrict__ valid,
    float* __restrict__ tb, int* __restrict__ tv)
{
  const int t = blockIdx.x * 256 + threadIdx.x;
  if (t >= PRE_N) return;
  const unsigned idx = 0xFFFFFFFFu - (unsigned)(keys[t] & 0xFFFFFFFFu);
  tb[t * 4 + 0] = boxes[idx * 4 + 0];
  tb[t * 4 + 1] = boxes[idx * 4 + 1];
  tb[t * 4 + 2] = boxes[idx * 4 + 2];
  tb[t * 4 + 3] = boxes[idx * 4 + 3];
  tv[t] = valid[idx];
}

// Kernel 6: IoU suppression bitmask. Thread (i, wj) -> 32 IoUs -> one word.
__global__ __launch_bounds__(256) void rpn_iou_mask(
    const float* __restrict__ tb, unsigned* __restrict__ mask)
{
  const int t = blockIdx.x * 256 + threadIdx.x;
  if (t >= PRE_N * NMS_WORDS) return;
  const int i  = t / NMS_WORDS;
  const int wj = t - i * NMS_WORDS;

  const float x1 = tb[i * 4 + 0], y1 = tb[i * 4 + 1];
  const float x2 = tb[i * 4 + 2], y2 = tb[i * 4 + 3];
  const float ai = (x2 - x1 + 1.f) * (y2 - y1 + 1.f);

  unsigned bits = 0u;
  const int j0 = wj * 32;
  for (int b = 0; b < 32; ++b) {
    const int j = j0 + b;
    if (j <= i || j >= PRE_N) continue;
    const float bx1 = tb[j * 4 + 0], by1 = tb[j * 4 + 1];
    const float bx2 = tb[j * 4 + 2], by2 = tb[j * 4 + 3];
    const float iw = fminf(x2, bx2) - fmaxf(x1, bx1) + 1.f;
    const float ih = fminf(y2, by2) - fmaxf(y1, by1) + 1.f;
    const float inter = fmaxf(iw, 0.f) * fmaxf(ih, 0.f);
    const float aj = (bx2 - bx1 + 1.f) * (by2 - by1 + 1.f);
    const float iou = inter / (ai + aj - inter);
    if (iou > 0.7f) bits |= 1u << b;
  }
  mask[t] = bits;
}

// Kernel 7: sequential NMS scan (single block) + ROI writeout.
__global__ __launch_bounds__(256) void rpn_nms_scan(
    const float* __restrict__ tb, const int* __restrict__ tv,
    const unsigned* __restrict__ mask, float* __restrict__ out)
{
  __shared__ unsigned removed[NMS_WORDS];
  __shared__ int keep[POST_N];
  __shared__ int cnt;
  __shared__ int alive;

  const int t = threadIdx.x;
  for (int w = t; w < NMS_WORDS; w += 256) removed[w] = 0u;
  if (t == 0) cnt = 0;
  __syncthreads();

  for (int i = 0; i < PRE_N; ++i) {
    if (t == 0) {
      const bool sup = (tv[i] == 0) || ((removed[i >> 5] >> (i & 31)) & 1u);
      alive = sup ? 0 : 1;
      if (!sup) { if (cnt < POST_N) keep[cnt] = i; cnt++; }
    }
    __syncthreads();
    if (alive) {
      for (int w = t; w < NMS_WORDS; w += 256)
        removed[w] |= mask[(size_t)i * NMS_WORDS + w];
    }
    __syncthreads();
  }

  for (int r = t; r < POST_N; r += 256) {
    float x1 = 0.f, y1 = 0.f, x2 = 0.f, y2 = 0.f;
    if (r < cnt) {
      const int i = keep[r];
      x1 = tb[i * 4 + 0]; y1 = tb[i * 4 + 1];
      x2 = tb[i * 4 + 2]; y2 = tb[i * 4 + 3];
    }
    out[r * 5 + 0] = 0.f;
    out[r * 5 + 1] = x1; out[r * 5 + 2] = y1;
    out[r * 5 + 3] = x2; out[r * 5 + 4] = y2;
  }
}

// ---------------------------------------------------------------------------
// Host: base-anchor generation (double math; rint == round-half-even matches
// np.round), workspace carving, launches.
// ---------------------------------------------------------------------------
static void rpn_gen_anchors(float* out /*[36]*/) {
  const double ratios[3] = {0.5, 1.0, 2.0};
  const double scales[3] = {8.0, 16.0, 32.0};
  const double bw = 16.0, bh = 16.0;
  const double xc = 0.0 + 0.5 * (bw - 1.0);
  const double yc = 0.0 + 0.5 * (bh - 1.0);
  const double size = bw * bh;
  double ra[3][4];
  for (int i = 0; i < 3; ++i) {
    const double ws = rint(sqrt(size / ratios[i]));
    const double hs = rint(ws * ratios[i]);
    ra[i][0] = xc - 0.5 * (ws - 1.0); ra[i][1] = yc - 0.5 * (hs - 1.0);
    ra[i][2] = xc + 0.5 * (ws - 1.0); ra[i][3] = yc + 0.5 * (hs - 1.0);
  }
  int o = 0;
  for (int i = 0; i < 3; ++i) {
    const double w2 = ra[i][2] - ra[i][0] + 1.0;
    const double h2 = ra[i][3] - ra[i][1] + 1.0;
    const double xc2 = ra[i][0] + 0.5 * (w2 - 1.0);
    const double yc2 = ra[i][1] + 0.5 * (h2 - 1.0);
    for (int s = 0; s < 3; ++s) {
      const double ws = w2 * scales[s], hs = h2 * scales[s];
      out[o * 4 + 0] = (float)(xc2 - 0.5 * (ws - 1.0));
      out[o * 4 + 1] = (float)(yc2 - 0.5 * (hs - 1.0));
      out[o * 4 + 2] = (float)(xc2 + 0.5 * (ws - 1.0));
      out[o * 4 + 3] = (float)(yc2 + 0.5 * (hs - 1.0));
      ++o;
    }
  }
}

extern "C" void kernel_launch(void* const* d_in, const int* in_sizes, int n_in,
                              void* d_out, int out_size, void* d_ws, size_t ws_size,
                              hipStream_t stream) {
  const float* feature = (const float*)d_in[0];
  // d_in[1] = gt_boxes (unused by reference output path)
  const float* im_info = (const float*)d_in[2];
  const float* conv_w  = (const float*)d_in[3];
  const float* conv_b  = (const float*)d_in[4];
  const float* cls_w   = (const float*)d_in[5];
  const float* cls_b   = (const float*)d_in[6];
  const float* reg_w   = (const float*)d_in[7];
  const float* reg_b   = (const float*)d_in[8];

  char* ws = (char*)d_ws;
  float* feats = (float*)ws;                 ws += (size_t)RPN_M * RPN_OC * 4;   // 13.1 MB
  float* head  = (float*)ws;                 ws += (size_t)RPN_M * 64 * 4;       // 1.64 MB
  float* boxes = (float*)ws;                 ws += (size_t)NBOX * 4 * 4;         // 0.92 MB
  int*   valid = (int*)ws;                   ws += (size_t)NBOX * 4;             // 0.23 MB
  unsigned long long* keys = (unsigned long long*)ws; ws += (size_t)NPAD * 8;    // 0.52 MB
  float* tb    = (float*)ws;                 ws += (size_t)PRE_N * 4 * 4;        // 96 KB
  int*   tv    = (int*)ws;                   ws += (size_t)PRE_N * 4;            // 24 KB
  unsigned* mask = (unsigned*)ws;            ws += (size_t)PRE_N * NMS_WORDS * 4;// 4.5 MB
  (void)ws_size; (void)in_sizes; (void)n_in; (void)out_size;

  Anchors an;
  rpn_gen_anchors(an.a);

  // 1) 3x3 conv + ReLU  (compute-bound; fp32 WMMA, ~60.4 GFLOP,
  //    async-to-LDS weight staging, double-buffered K pipeline)
  rpn_conv3x3_wmma<<<dim3(RPN_M / 64, RPN_OC / 64), 256, 0, stream>>>(
      feature, conv_w, conv_b, feats);

  // 2) fused 1x1 cls+reg heads
  rpn_heads_wmma<<<RPN_M / 32, 256, 0, stream>>>(
      feats, cls_w, cls_b, reg_w, reg_b, head);

  // 3) decode proposals + sort keys (pads [57600,65536) with key 0)
  rpn_proposals<<<NPAD / 256, 256, 0, stream>>>(
      head, im_info, an, boxes, valid, keys);

  // 4) full bitonic sort, descending (stable-argsort-equivalent keys)
  for (int k = 2; k <= NPAD; k <<= 1)
    for (int j = k >> 1; j > 0; j >>= 1)
      rpn_bitonic<<<NPAD / 256, 256, 0, stream>>>(keys, j, k);

  // 5) top-6000 gather
  rpn_gather_top<<<(PRE_N + 255) / 256, 256, 0, stream>>>(keys, boxes, valid, tb, tv);

  // 6) IoU bitmask
  rpn_iou_mask<<<(PRE_N * NMS_WORDS + 255) / 256, 256, 0, stream>>>(tb, mask);

  // 7) sequential NMS scan + write 300x5 ROIs
  rpn_nms_scan<<<1, 256, 0, stream>>>(tb, tv, mask, (float*)d_out);
}